// Graph_1047972020267
// MI455X (gfx1250) — compile-verified
//
#include <hip/hip_runtime.h>
#include <stdint.h>

#ifndef __has_builtin
#define __has_builtin(x) 0
#endif

#if __has_builtin(__builtin_amdgcn_global_load_async_to_lds_b128)
#define HAVE_ASYNC_LOAD 1
#else
#define HAVE_ASYNC_LOAD 0
#endif
#if __has_builtin(__builtin_amdgcn_global_store_async_from_lds_b128)
#define HAVE_ASYNC_STORE 1
#else
#define HAVE_ASYNC_STORE 0
#endif

// 16-byte payload type the async-LDS builtins expect (v4i32).
typedef int v4i __attribute__((__vector_size__(16)));
typedef __attribute__((address_space(1))) v4i g_v4i;   // global
typedef __attribute__((address_space(3))) v4i l_v4i;   // LDS

#define IMG_H 1024
#define IMG_W 1024

// ---- CDNA5 async global -> LDS copy (16 bytes per lane, ASYNCcnt-tracked) ----
__device__ __forceinline__ void async_g2l_b128(const float* g, float* lds_generic) {
#if HAVE_ASYNC_LOAD
  __builtin_amdgcn_global_load_async_to_lds_b128(
      (g_v4i*)(uintptr_t)(const void*)g, (l_v4i*)lds_generic, 0, 0);
#else
  uint32_t l = (uint32_t)(uintptr_t)(l_v4i*)lds_generic;
  asm volatile("global_load_async_to_lds_b128 %0, %1, off"
               :: "v"(l), "v"(g) : "memory");
#endif
}

// ---- CDNA5 async LDS -> global copy (16 bytes per lane, ASYNCcnt-tracked) ----
__device__ __forceinline__ void async_l2g_b128(float* g, const float* lds_generic) {
#if HAVE_ASYNC_STORE
  __builtin_amdgcn_global_store_async_from_lds_b128(
      (g_v4i*)(uintptr_t)(void*)g, (l_v4i*)lds_generic, 0, 0);
#else
  uint32_t l = (uint32_t)(uintptr_t)(l_v4i*)lds_generic;
  asm volatile("global_store_async_from_lds_b128 %0, %1, off"
               :: "v"(g), "v"(l) : "memory");
#endif
}

__device__ __forceinline__ void wait_async0() {
#if __has_builtin(__builtin_amdgcn_s_wait_asynccnt)
  __builtin_amdgcn_s_wait_asynccnt(0);
#else
  asm volatile("s_wait_asynccnt 0" ::: "memory");
#endif
}

// One workgroup per image row (b, j). 256 threads x 4 pixels each.
__global__ void __launch_bounds__(256)
stencil5_rows(const float* __restrict__ x, float* __restrict__ out) {
  __shared__ __align__(16) float s_in[3 * IMG_W];   // rows: up | center | down
  __shared__ __align__(16) float s_out[5 * IMG_W];  // interleaved [W][5]

  const int tid = threadIdx.x;
  const int row = blockIdx.x & (IMG_H - 1);
  const int b   = blockIdx.x >> 10;

  const int rup = (row > 0)         ? row - 1 : row;   // edge -> replicate center row
  const int rdn = (row < IMG_H - 1) ? row + 1 : row;

  const float* base = x + (size_t)b * ((size_t)IMG_H * IMG_W);
  const int k0 = tid * 4;

  // Phase 1: async DMA the three input rows into LDS (3 x b128 per lane).
  async_g2l_b128(base + (size_t)rup * IMG_W + k0, &s_in[0 * IMG_W + k0]);
  async_g2l_b128(base + (size_t)row * IMG_W + k0, &s_in[1 * IMG_W + k0]);
  async_g2l_b128(base + (size_t)rdn * IMG_W + k0, &s_in[2 * IMG_W + k0]);
  wait_async0();          // this wave's async loads landed in LDS
  __syncthreads();        // all waves' loads landed

  // Phase 2: gather 5 channels for 4 consecutive pixels; stage interleaved in LDS.
  float4 c = *(const float4*)&s_in[1 * IMG_W + k0];
  float4 u = *(const float4*)&s_in[0 * IMG_W + k0];
  float4 d = *(const float4*)&s_in[2 * IMG_W + k0];
  float prevv = (k0 == 0)           ? c.x : s_in[1 * IMG_W + k0 - 1];
  float nextv = (k0 + 4 >= IMG_W)   ? c.w : s_in[1 * IMG_W + k0 + 4];

  // per pixel order: [center, up, right, down, left]
  // 20 contiguous floats at byte offset tid*80 (16B aligned) -> 5 ds_store_b128
  float4* ob = (float4*)&s_out[k0 * 5];
  ob[0] = make_float4(c.x,  u.x,  c.y,  d.x);   // p0: c,u,r,d
  ob[1] = make_float4(prevv, c.y, u.y,  c.z);   // p0: l | p1: c,u,r
  ob[2] = make_float4(d.y,  c.x,  c.z,  u.z);   // p1: d,l | p2: c,u
  ob[3] = make_float4(c.w,  d.z,  c.y,  c.w);   // p2: r,d,l | p3: c
  ob[4] = make_float4(u.w,  nextv, d.w, c.z);   // p3: u,r,d,l

  __syncthreads();        // all s_out writes visible before DMA-out

  // Phase 3: async DMA the 20 KiB interleaved row out, fully coalesced b128.
  float* orow = out + (size_t)blockIdx.x * (5 * (size_t)IMG_W);
#pragma unroll
  for (int t = 0; t < 5; ++t) {
    const int i = tid + t * 256;          // 1280 x 16B chunks total
    async_l2g_b128(orow + i * 4, &s_out[i * 4]);
  }
  wait_async0();  // explicit drain (S_ENDPGM would also wait-idle)
}

extern "C" void kernel_launch(void* const* d_in, const int* in_sizes, int n_in,
                              void* d_out, int out_size, void* d_ws, size_t ws_size,
                              hipStream_t stream) {
  (void)n_in; (void)d_ws; (void)ws_size; (void)out_size;
  const float* x = (const float*)d_in[0];
  float* out = (float*)d_out;
  const int B = in_sizes[0] / (IMG_H * IMG_W);   // = 16
  dim3 grid((unsigned)(B * IMG_H));
  stencil5_rows<<<grid, 256, 0, stream>>>(x, out);
}